// GHMIoU_1580547973496
// MI455X (gfx1250) — compile-verified
//
#include <hip/hip_runtime.h>

#define GBINS   10
#define BLOCKT  256
#define WAVE    32
#define VEC     4
#define TILE    (BLOCKT * VEC)   // 1024 elements per tile per block

typedef float v4f __attribute__((ext_vector_type(4)));
typedef int   v4i __attribute__((vector_size(4 * sizeof(int))));

#define AS1 __attribute__((address_space(1)))
#define AS3 __attribute__((address_space(3)))

// CPol: TH in [2:0] (1 = non-temporal), scope in [4:3] (0 = WGP)
#define CPOL_NT 1

// Prefer the CDNA5 async global->LDS streaming path when the builtin exists.
#if defined(__HIP_DEVICE_COMPILE__) && __has_builtin(__builtin_amdgcn_global_load_async_to_lds_b128) && __has_builtin(__builtin_amdgcn_s_wait_asynccnt)
#define GHM_ASYNC 1
#else
#define GHM_ASYNC 0
#endif

#if GHM_ASYNC
__device__ __forceinline__ void async_b128(const float* g, float* l) {
    __builtin_amdgcn_global_load_async_to_lds_b128(
        (AS1 v4i*)(g), (AS3 v4i*)(l), 0, CPOL_NT);
}
#endif

// Per-element: sigmoid/bin/BCE, accumulate into bank-conflict-free LDS histogram.
// Layout hist[bin*32 + lane]: 32 lanes -> 32 distinct banks, one ds_add_f32
// per wave retires 32 updates. Atomic handles inter-wave sharing of a column.
__device__ __forceinline__ void ghm_accum(float p, float t, float w,
                                          float* s_cnt, float* s_bce, int lane) {
    if (!(w > 0.0f)) return;                       // label_weight > 0
    float ap  = __builtin_fabsf(p);
    float e   = __expf(-ap);                       // shared by sigmoid & BCE
    float inv = __builtin_amdgcn_rcpf(1.0f + e);
    float sg  = (p >= 0.0f) ? inv : e * inv;       // stable sigmoid
    float g   = __builtin_fabsf(sg - t);           // in [0,1]
    int   b   = (int)(g * (float)GBINS);
    b = (b > GBINS - 1) ? (GBINS - 1) : b;
    // logaddexp(0,p) - p*t = max(p,0) + log(1+exp(-|p|)) - p*t
    float bce = __builtin_fmaxf(p, 0.0f) + __logf(1.0f + e) - p * t;
    atomicAdd(&s_cnt[b * WAVE + lane], 1.0f);      // ds_add_f32
    atomicAdd(&s_bce[b * WAVE + lane], bce);       // ds_add_f32
}

__global__ void ghm_zero(float* ws, int count) {
    int i = blockIdx.x * blockDim.x + threadIdx.x;
    if (i < count) ws[i] = 0.0f;
}

// ws layout: counts at ws[b*stripes + s], bce sums at ws[(GBINS+b)*stripes + s]
__global__ __launch_bounds__(BLOCKT)
void ghm_hist(const float* __restrict__ pred,
              const float* __restrict__ targ,
              const float* __restrict__ lw,
              float* __restrict__ ws, int n, int stripes) {
    __shared__ float s_cnt[GBINS * WAVE];
    __shared__ float s_bce[GBINS * WAVE];
    const int tid  = threadIdx.x;
    const int lane = tid & (WAVE - 1);

    for (int i = tid; i < GBINS * WAVE; i += BLOCKT) { s_cnt[i] = 0.0f; s_bce[i] = 0.0f; }
    __syncthreads();

    const int num_tiles = n / TILE;

#if GHM_ASYNC
    // Self-staged per-lane double buffer: each lane async-loads exactly the
    // 16B it will consume, so only s_wait_asynccnt is needed (no barrier).
    __shared__ float s_p[2][TILE];
    __shared__ float s_t[2][TILE];
    __shared__ float s_w[2][TILE];

    int tile = blockIdx.x;
    int buf  = 0;
    if (tile < num_tiles) {
        int base = tile * TILE + tid * VEC;
        async_b128(pred + base, &s_p[0][tid * VEC]);
        async_b128(targ + base, &s_t[0][tid * VEC]);
        async_b128(lw   + base, &s_w[0][tid * VEC]);
    }
    for (; tile < num_tiles; tile += gridDim.x) {
        int nxt = tile + gridDim.x;
        if (nxt < num_tiles) {
            int nb = nxt * TILE + tid * VEC;
            async_b128(pred + nb, &s_p[buf ^ 1][tid * VEC]);
            async_b128(targ + nb, &s_t[buf ^ 1][tid * VEC]);
            async_b128(lw   + nb, &s_w[buf ^ 1][tid * VEC]);
            __builtin_amdgcn_s_wait_asynccnt(3);   // current 3 done, next 3 in flight
        } else {
            __builtin_amdgcn_s_wait_asynccnt(0);
        }
        v4f p4 = *(const v4f*)&s_p[buf][tid * VEC];   // ds_load_b128
        v4f t4 = *(const v4f*)&s_t[buf][tid * VEC];
        v4f w4 = *(const v4f*)&s_w[buf][tid * VEC];
        ghm_accum(p4.x, t4.x, w4.x, s_cnt, s_bce, lane);
        ghm_accum(p4.y, t4.y, w4.y, s_cnt, s_bce, lane);
        ghm_accum(p4.z, t4.z, w4.z, s_cnt, s_bce, lane);
        ghm_accum(p4.w, t4.w, w4.w, s_cnt, s_bce, lane);
        buf ^= 1;
    }
#else
    for (int tile = blockIdx.x; tile < num_tiles; tile += gridDim.x) {
        int base = tile * TILE + tid * VEC;
        int nb   = base + (int)gridDim.x * TILE;
        if (nb + VEC <= n) {                         // global_prefetch_b8 next tile
            __builtin_prefetch(pred + nb, 0, 0);
            __builtin_prefetch(targ + nb, 0, 0);
            __builtin_prefetch(lw + nb, 0, 0);
        }
        v4f p4 = __builtin_nontemporal_load((const v4f*)(pred + base)); // b128 NT
        v4f t4 = __builtin_nontemporal_load((const v4f*)(targ + base));
        v4f w4 = __builtin_nontemporal_load((const v4f*)(lw + base));
        ghm_accum(p4.x, t4.x, w4.x, s_cnt, s_bce, lane);
        ghm_accum(p4.y, t4.y, w4.y, s_cnt, s_bce, lane);
        ghm_accum(p4.z, t4.z, w4.z, s_cnt, s_bce, lane);
        ghm_accum(p4.w, t4.w, w4.w, s_cnt, s_bce, lane);
    }
#endif

    // tail (n % TILE), handled by block 0 only
    if (blockIdx.x == 0) {
        for (int i = num_tiles * TILE + tid; i < n; i += BLOCKT)
            ghm_accum(pred[i], targ[i], lw[i], s_cnt, s_bce, lane);
    }

    __syncthreads();
    if (tid < GBINS) {
        float c = 0.0f, s = 0.0f;
        for (int l = 0; l < WAVE; ++l) {
            c += s_cnt[tid * WAVE + l];
            s += s_bce[tid * WAVE + l];
        }
        int sl = (int)(blockIdx.x % (unsigned)stripes);  // spread same-address atomics
        atomicAdd(&ws[tid * stripes + sl], c);           // global_atomic_add_f32
        atomicAdd(&ws[(GBINS + tid) * stripes + sl], s);
    }
}

// One wave per bin (block = 320 threads): parallel stripe gather + wave
// reduction, then a single-thread 10-element combine.
__global__ __launch_bounds__(GBINS * WAVE)
void ghm_final(const float* __restrict__ ws, float* __restrict__ out, int stripes) {
    __shared__ float s_ratio[GBINS];
    __shared__ float s_flag[GBINS];
    const int b = threadIdx.x / WAVE;   // bin handled by this wave
    const int l = threadIdx.x % WAVE;

    float c = 0.0f, s = 0.0f;
    for (int j = l; j < stripes; j += WAVE) {
        c += ws[b * stripes + j];
        s += ws[(GBINS + b) * stripes + j];
    }
    for (int off = WAVE / 2; off > 0; off >>= 1) {   // wave32 shuffle reduce
        c += __shfl_down(c, off, WAVE);
        s += __shfl_down(s, off, WAVE);
    }
    if (l == 0) {
        s_ratio[b] = (c > 0.0f) ? s / c : 0.0f;
        s_flag[b]  = (c > 0.0f) ? 1.0f : 0.0f;
    }
    __syncthreads();
    if (threadIdx.x == 0) {
        float acc = 0.0f, nb = 0.0f;
        for (int i = 0; i < GBINS; ++i) { acc += s_ratio[i]; nb += s_flag[i]; }
        out[0] = acc / __builtin_fmaxf(nb, 1.0f);   // * LOSS_WEIGHT (1.0); tot cancels
    }
}

extern "C" void kernel_launch(void* const* d_in, const int* in_sizes, int n_in,
                              void* d_out, int out_size, void* d_ws, size_t ws_size,
                              hipStream_t stream) {
    const float* pred = (const float*)d_in[0];
    const float* targ = (const float*)d_in[1];
    const float* lw   = (const float*)d_in[2];
    // d_in[3] is bins (=10); fixed at compile time per the reference.
    float* out = (float*)d_out;
    float* ws  = (float*)d_ws;
    int n = in_sizes[0];

    int stripes = 64;
    while (stripes > 1 && (size_t)(2 * GBINS * stripes) * sizeof(float) > ws_size)
        stripes >>= 1;

    int zc = 2 * GBINS * stripes;
    ghm_zero<<<(zc + 255) / 256, 256, 0, stream>>>(ws, zc);

    int num_tiles = n / TILE;
    int grid = num_tiles < 4096 ? (num_tiles > 0 ? num_tiles : 1) : 4096;
    ghm_hist<<<grid, BLOCKT, 0, stream>>>(pred, targ, lw, ws, n, stripes);

    ghm_final<<<1, GBINS * WAVE, 0, stream>>>(ws, out, stripes);
}